// GCN_73452530696965
// MI455X (gfx1250) — compile-verified
//
#include <hip/hip_runtime.h>
#include <math.h>

typedef __attribute__((ext_vector_type(2))) float v2f;
typedef __attribute__((ext_vector_type(8))) float v8f;

#define NN   50000
#define NE   1600000
#define FIN  512
#define FH   128
#define FOUT 40
#define FP   64          // W2 padded column count (4 full 16-wide N tiles)

// ---------------------------------------------------------------------------
// Degree / normalization
// ---------------------------------------------------------------------------
__global__ __launch_bounds__(256) void deg_init_kernel(float* __restrict__ deg) {
    int i = blockIdx.x * 256 + threadIdx.x;
    if (i < NN) deg[i] = 1.0f;                       // self-loop
}

__global__ __launch_bounds__(256) void edge_deg_kernel(const int* __restrict__ dst,
                                                       float* __restrict__ deg) {
    int e = blockIdx.x * 256 + threadIdx.x;
    if (e < NE) atomicAdd(&deg[dst[e]], 1.0f);
}

__global__ __launch_bounds__(256) void dinv_kernel(float* __restrict__ deg) {
    int i = blockIdx.x * 256 + threadIdx.x;
    if (i < NN) {
        float d = deg[i];
        deg[i] = (d > 0.0f) ? rsqrtf(d) : 0.0f;      // in place: deg -> dinv
    }
}

// ---------------------------------------------------------------------------
// W2 zero-pad to [FH x FP]
// ---------------------------------------------------------------------------
__global__ __launch_bounds__(256) void pad_w2_kernel(const float* __restrict__ W2,
                                                     float* __restrict__ W2p) {
    int g = blockIdx.x * 256 + threadIdx.x;          // 128*64 = 8192 threads
    if (g < FH * FP) {
        int r = g >> 6, c = g & 63;
        W2p[g] = (c < FOUT) ? W2[r * FOUT + c] : 0.0f;
    }
}

// ---------------------------------------------------------------------------
// GEMM1: H1[50000x128] = X[50000x512] * W1[512x128]   (fp32 WMMA 16x16x4)
// block = 256 threads (8 waves), one 16-row tile per block, one 16-col tile
// per wave. A tile staged in LDS with stride 516 (bank-conflict-free).
// ---------------------------------------------------------------------------
__global__ __launch_bounds__(256) void gemm1_kernel(const float* __restrict__ X,
                                                    const float* __restrict__ W,
                                                    float* __restrict__ H) {
    __shared__ float As[16 * 516];
    const int row0 = blockIdx.x * 16;
    const int tid  = threadIdx.x;

    // stage 16x512 fp32 tile: 2048 float4, 8 per thread
#pragma unroll
    for (int j = 0; j < 8; ++j) {
        int q = tid + 256 * j;                       // 0..2047
        int r = q >> 7;                              // 128 float4 per row
        int c = q & 127;
        float4 v = *(const float4*)&X[(row0 + r) * FIN + c * 4];
        float* p = &As[r * 516 + c * 4];
        p[0] = v.x; p[1] = v.y; p[2] = v.z; p[3] = v.w;
    }
    __syncthreads();

    const int wave = tid >> 5;                       // 0..7 -> N tile
    const int lane = tid & 31;
    const int m    = lane & 15;
    const int hi   = lane >> 4;                      // 0: K,K+1   1: K+2,K+3
    const int n    = (wave << 4) + m;

    v8f acc = {};
#pragma unroll 4
    for (int k = 0; k < FIN; k += 4) {
        const float2 av = *(const float2*)&As[m * 516 + k + 2 * hi];
        v2f a, b;
        a[0] = av.x;                                 // A[m][k+2*hi]
        a[1] = av.y;                                 // A[m][k+1+2*hi]
        b[0] = W[(k + 2 * hi) * FH + n];             // B[k+2*hi][n]
        b[1] = W[(k + 1 + 2 * hi) * FH + n];         // B[k+1+2*hi][n]
        acc = __builtin_amdgcn_wmma_f32_16x16x4_f32(false, a, false, b,
                                                    (short)0, acc, false, false);
    }
#pragma unroll
    for (int v = 0; v < 8; ++v)
        H[(row0 + v + 8 * hi) * FH + n] = acc[v];
}

// ---------------------------------------------------------------------------
// GEMM2: H2[50000x40] = Hrelu[50000x128] * W2p[128x64]
// block = 128 threads (4 waves), all waves run WMMA (EXEC all-1s), store n<40
// ---------------------------------------------------------------------------
__global__ __launch_bounds__(128) void gemm2_kernel(const float* __restrict__ A,
                                                    const float* __restrict__ W2p,
                                                    float* __restrict__ H2) {
    __shared__ float As[16 * 132];
    const int row0 = blockIdx.x * 16;
    const int tid  = threadIdx.x;

    // stage 16x128 fp32 tile: 512 float4, 4 per thread
#pragma unroll
    for (int j = 0; j < 4; ++j) {
        int q = tid + 128 * j;                       // 0..511
        int r = q >> 5;                              // 32 float4 per row
        int c = q & 31;
        float4 v = *(const float4*)&A[(row0 + r) * FH + c * 4];
        float* p = &As[r * 132 + c * 4];
        p[0] = v.x; p[1] = v.y; p[2] = v.z; p[3] = v.w;
    }
    __syncthreads();

    const int wave = tid >> 5;                       // 0..3 -> N tile
    const int lane = tid & 31;
    const int m    = lane & 15;
    const int hi   = lane >> 4;
    const int n    = (wave << 4) + m;

    v8f acc = {};
#pragma unroll 4
    for (int k = 0; k < FH; k += 4) {
        const float2 av = *(const float2*)&As[m * 132 + k + 2 * hi];
        v2f a, b;
        a[0] = av.x;
        a[1] = av.y;
        b[0] = W2p[(k + 2 * hi) * FP + n];
        b[1] = W2p[(k + 1 + 2 * hi) * FP + n];
        acc = __builtin_amdgcn_wmma_f32_16x16x4_f32(false, a, false, b,
                                                    (short)0, acc, false, false);
    }
    if (n < FOUT) {
#pragma unroll
        for (int v = 0; v < 8; ++v)
            H2[(row0 + v + 8 * hi) * FOUT + n] = acc[v];
    }
}

// ---------------------------------------------------------------------------
// Aggregation layer 1 (128 features)
// ---------------------------------------------------------------------------
__global__ __launch_bounds__(256) void agg1_init_kernel(const float* __restrict__ h1,
                                                        const float* __restrict__ dinv,
                                                        float* __restrict__ agg) {
    int g = blockIdx.x * 256 + threadIdx.x;          // NN*32 threads
    int i = g >> 5, c = g & 31;
    float s = dinv[i] * dinv[i];                     // self-loop norm
    float4 v = *(const float4*)&h1[i * FH + c * 4];
    float4 o; o.x = v.x * s; o.y = v.y * s; o.z = v.z * s; o.w = v.w * s;
    *(float4*)&agg[i * FH + c * 4] = o;
}

__global__ __launch_bounds__(256) void agg1_edge_kernel(const int* __restrict__ src,
                                                        const int* __restrict__ dst,
                                                        const float* __restrict__ dinv,
                                                        const float* __restrict__ h1,
                                                        float* __restrict__ agg) {
    int g = blockIdx.x * 256 + threadIdx.x;          // NE*32 threads (exact)
    int e = g >> 5, lane = g & 31;
    int s = src[e], d = dst[e];
    float nrm = dinv[s] * dinv[d];
    float4 v = *(const float4*)&h1[s * FH + lane * 4];
    float* o = &agg[d * FH + lane * 4];
    atomicAdd(o + 0, v.x * nrm);
    atomicAdd(o + 1, v.y * nrm);
    atomicAdd(o + 2, v.z * nrm);
    atomicAdd(o + 3, v.w * nrm);
}

__global__ __launch_bounds__(256) void relu_bias_kernel(float* __restrict__ agg,
                                                        const float* __restrict__ b1) {
    int g = blockIdx.x * 256 + threadIdx.x;          // NN*32 threads
    int c = g & 31;
    float4 b = ((const float4*)b1)[c];
    float4 v = ((float4*)agg)[g];
    v.x = fmaxf(v.x + b.x, 0.0f);
    v.y = fmaxf(v.y + b.y, 0.0f);
    v.z = fmaxf(v.z + b.z, 0.0f);
    v.w = fmaxf(v.w + b.w, 0.0f);
    ((float4*)agg)[g] = v;
}

// ---------------------------------------------------------------------------
// Aggregation layer 2 (40 features -> 10 float4 chunks)
// ---------------------------------------------------------------------------
__global__ __launch_bounds__(256) void agg2_init_kernel(const float* __restrict__ h2,
                                                        const float* __restrict__ dinv,
                                                        float* __restrict__ agg) {
    int g = blockIdx.x * 256 + threadIdx.x;
    if (g >= NN * 10) return;
    int i = g / 10, c = g % 10;
    float s = dinv[i] * dinv[i];
    float4 v = *(const float4*)&h2[i * FOUT + c * 4];
    float4 o; o.x = v.x * s; o.y = v.y * s; o.z = v.z * s; o.w = v.w * s;
    *(float4*)&agg[i * FOUT + c * 4] = o;
}

__global__ __launch_bounds__(256) void agg2_edge_kernel(const int* __restrict__ src,
                                                        const int* __restrict__ dst,
                                                        const float* __restrict__ dinv,
                                                        const float* __restrict__ h2,
                                                        float* __restrict__ agg) {
    int g = blockIdx.x * 256 + threadIdx.x;
    if (g >= NE * 10) return;
    int e = g / 10, c = g % 10;
    int s = src[e], d = dst[e];
    float nrm = dinv[s] * dinv[d];
    float4 v = *(const float4*)&h2[s * FOUT + c * 4];
    float* o = &agg[d * FOUT + c * 4];
    atomicAdd(o + 0, v.x * nrm);
    atomicAdd(o + 1, v.y * nrm);
    atomicAdd(o + 2, v.z * nrm);
    atomicAdd(o + 3, v.w * nrm);
}

// ---------------------------------------------------------------------------
// Bias + log_softmax over 40 classes: one wave32 per node (8 nodes / block)
// ---------------------------------------------------------------------------
__global__ __launch_bounds__(256) void logsoftmax_kernel(const float* __restrict__ agg,
                                                         const float* __restrict__ b2,
                                                         float* __restrict__ out) {
    int wave = threadIdx.x >> 5;
    int lane = threadIdx.x & 31;
    int i = blockIdx.x * 8 + wave;                   // exact: 6250*8 = NN

    float v1 = agg[i * FOUT + lane] + b2[lane];      // lane 0..31 < 40
    float v2 = -INFINITY;
    if (lane < 8) v2 = agg[i * FOUT + 32 + lane] + b2[32 + lane];

    float m = fmaxf(v1, v2);
#pragma unroll
    for (int o = 16; o >= 1; o >>= 1) m = fmaxf(m, __shfl_xor(m, o, 32));

    float s = expf(v1 - m) + ((lane < 8) ? expf(v2 - m) : 0.0f);
#pragma unroll
    for (int o = 16; o >= 1; o >>= 1) s += __shfl_xor(s, o, 32);

    float l = m + logf(s);
    out[i * FOUT + lane] = v1 - l;
    if (lane < 8) out[i * FOUT + 32 + lane] = v2 - l;
}

// ---------------------------------------------------------------------------
// Host launcher
// ---------------------------------------------------------------------------
extern "C" void kernel_launch(void* const* d_in, const int* in_sizes, int n_in,
                              void* d_out, int out_size, void* d_ws, size_t ws_size,
                              hipStream_t stream) {
    const float* x    = (const float*)d_in[0];       // [NN, FIN]
    const int*   ei   = (const int*)d_in[1];         // [2, NE]
    const float* W1   = (const float*)d_in[2];       // [FIN, FH]
    const float* b1   = (const float*)d_in[3];       // [FH]
    const float* W2   = (const float*)d_in[4];       // [FH, FOUT]
    const float* b2   = (const float*)d_in[5];       // [FOUT]
    float*       out  = (float*)d_out;               // [NN, FOUT]

    const int* src = ei;
    const int* dst = ei + NE;

    // Workspace layout (floats, 64-float aligned blocks)
    float* ws       = (float*)d_ws;
    float* w_dinv   = ws;                            // 50048
    float* w_h1     = w_dinv + 50048;                // NN*FH   = 6,400,000
    float* w_agg1   = w_h1   + NN * FH;              // NN*FH   = 6,400,000
    float* w_h2     = w_agg1 + NN * FH;              // NN*FOUT = 2,000,000
    float* w_agg2   = w_h2   + NN * FOUT;            // NN*FOUT = 2,000,000
    float* w_w2p    = w_agg2 + NN * FOUT;            // FH*FP   = 8,192

    // 1) degrees -> dinv (in place)
    deg_init_kernel<<<(NN + 255) / 256, 256, 0, stream>>>(w_dinv);
    edge_deg_kernel<<<(NE + 255) / 256, 256, 0, stream>>>(dst, w_dinv);
    dinv_kernel<<<(NN + 255) / 256, 256, 0, stream>>>(w_dinv);

    // 2) pad W2
    pad_w2_kernel<<<(FH * FP + 255) / 256, 256, 0, stream>>>(W2, w_w2p);

    // 3) h1 = x @ W1  (WMMA)
    gemm1_kernel<<<NN / 16, 256, 0, stream>>>(x, W1, w_h1);

    // 4) layer-1 aggregation: self-loop init + edge atomics, relu+bias
    agg1_init_kernel<<<NN * 32 / 256, 256, 0, stream>>>(w_h1, w_dinv, w_agg1);
    agg1_edge_kernel<<<NE * 32 / 256, 256, 0, stream>>>(src, dst, w_dinv, w_h1, w_agg1);
    relu_bias_kernel<<<NN * 32 / 256, 256, 0, stream>>>(w_agg1, b1);

    // 5) h2 = hrelu @ W2  (WMMA)
    gemm2_kernel<<<NN / 16, 128, 0, stream>>>(w_agg1, w_w2p, w_h2);

    // 6) layer-2 aggregation
    agg2_init_kernel<<<(NN * 10 + 255) / 256, 256, 0, stream>>>(w_h2, w_dinv, w_agg2);
    agg2_edge_kernel<<<(NE * 10 + 255) / 256, 256, 0, stream>>>(src, dst, w_dinv, w_h2, w_agg2);

    // 7) bias + log_softmax
    logsoftmax_kernel<<<NN / 8, 256, 0, stream>>>(w_agg2, b2, out);
}